// inference_41626823033167
// MI455X (gfx1250) — compile-verified
//
#include <hip/hip_runtime.h>
#include <hip/hip_bf16.h>

typedef float v2f __attribute__((ext_vector_type(2)));
typedef float v4f __attribute__((ext_vector_type(4)));
typedef float v8f __attribute__((ext_vector_type(8)));

#define N_NODES 4096
#define FDIM 30
#define HPAD 32
#define RREL 4
#define BK 128            // bias columns per chunk
#define TM 16             // rows per block tile
#define COEF_STRIDE 132   // padded LDS stride (conflict-free for strided row reads)

// ---------------- Kernel A: fts = x@W (padded, ones column), f1, f2 ----------------
__global__ void prep_kernel(const float* __restrict__ x,    // [N,30]
                            const float* __restrict__ W,    // [R,30,30]
                            const float* __restrict__ a1,   // [R,30]
                            const float* __restrict__ b1,   // [R]
                            const float* __restrict__ a2,   // [R,30]
                            const float* __restrict__ b2,   // [R]
                            float* __restrict__ ftsWs,      // [R,N,32]
                            float* __restrict__ f1Ws,       // [R,N]
                            float* __restrict__ f2Ws) {     // [R,N]
  int idx = blockIdx.x * blockDim.x + threadIdx.x;
  if (idx >= RREL * N_NODES) return;
  int r = idx / N_NODES;
  int n = idx - r * N_NODES;
  float xr[FDIM];
#pragma unroll
  for (int f = 0; f < FDIM; ++f) xr[f] = x[n * FDIM + f];
  const float* Wr  = W  + r * FDIM * FDIM;
  const float* a1r = a1 + r * FDIM;
  const float* a2r = a2 + r * FDIM;
  float f1 = b1[r], f2 = b2[r];
  float* out = ftsWs + ((size_t)r * N_NODES + n) * HPAD;
#pragma unroll 2
  for (int h = 0; h < FDIM; ++h) {
    float acc = 0.f;
#pragma unroll
    for (int f = 0; f < FDIM; ++f) acc = fmaf(xr[f], Wr[f * FDIM + h], acc);
    out[h] = acc;
    f1 = fmaf(acc, a1r[h], f1);
    f2 = fmaf(acc, a2r[h], f2);
  }
  out[30] = 1.0f;   // ones column -> WMMA accumulates softmax denominator for free
  out[31] = 0.0f;
  f1Ws[idx] = f1;
  f2Ws[idx] = f2;
}

// ---------------- Kernel B: fused masked-softmax attention, coefs@fts via WMMA ------
__global__ __launch_bounds__(256) void attn_kernel(
    const float* __restrict__ bias,    // [R,N,N] (0 or -1e9)
    const float* __restrict__ x,       // [N,30]
    const float* __restrict__ bz,      // [R,30]
    const float* __restrict__ ftsWs,   // [R,N,32]
    const float* __restrict__ f1Ws,    // [R,N]
    const float* __restrict__ f2Ws,    // [R,N]
    float* __restrict__ valsWs) {      // [R,N,32]
  __shared__ float f1s[TM];
  __shared__ float coefLds[TM * COEF_STRIDE];
  __shared__ float red[8][512];
  __shared__ float rD[TM][HPAD];

  const int r   = blockIdx.y;
  const int i0  = blockIdx.x * TM;
  const int tid = threadIdx.x;
  const int lane = tid & 31;
  const int w    = tid >> 5;      // wave id 0..7
  const int m16  = lane & 15;
  const int hi   = lane >> 4;

  if (tid < TM) f1s[tid] = f1Ws[r * N_NODES + i0 + tid];
  __syncthreads();

  const float* f2r     = f2Ws + r * N_NODES;
  const float* biasRow = bias + ((size_t)r * N_NODES + i0) * N_NODES;
  const float* ftsR    = ftsWs + (size_t)r * N_NODES * HPAD;

  v8f d0 = {};  // N = 0..15
  v8f d1 = {};  // N = 16..31 (col 30 = denominator)

  for (int j0 = 0; j0 < N_NODES; j0 += BK) {
    // --- cooperative: coef tile [TM][BK] -> LDS, fused exp(lrelu(f1+f2)+bias) ---
    {
      int row = tid >> 4;
      int q0  = tid & 15;
      float f1v = f1s[row];
      const v4f* bsrc = (const v4f*)(biasRow + (size_t)row * N_NODES + j0);
      const v4f* fsrc = (const v4f*)(f2r + j0);
#pragma unroll
      for (int s = 0; s < 2; ++s) {
        int q = q0 + s * 16;
        v4f b4 = __builtin_nontemporal_load(&bsrc[q]);  // NT: don't pollute L2
        v4f f2v = fsrc[q];
        v4f c; float t;
        t = f1v + f2v.x; t = fmaxf(t, 0.2f * t); c.x = __expf(t + b4.x);
        t = f1v + f2v.y; t = fmaxf(t, 0.2f * t); c.y = __expf(t + b4.y);
        t = f1v + f2v.z; t = fmaxf(t, 0.2f * t); c.z = __expf(t + b4.z);
        t = f1v + f2v.w; t = fmaxf(t, 0.2f * t); c.w = __expf(t + b4.w);
        *(v4f*)(&coefLds[row * COEF_STRIDE + 4 * q]) = c;
      }
    }
    __syncthreads();
    // --- wave w: WMMA over its 16-column stripe, K-steps of 4 ---
#pragma unroll
    for (int kk = 0; kk < 16; kk += 4) {
      int c0 = w * 16 + kk;
      v2f a;  // A 16x4: lane m = l%16; VGPR0 -> K=2*hi, VGPR1 -> K=2*hi+1
      a.x = coefLds[m16 * COEF_STRIDE + c0 + 2 * hi];
      a.y = coefLds[m16 * COEF_STRIDE + c0 + 2 * hi + 1];
      int jA = j0 + c0 + 2 * hi;
      const float* fb0 = ftsR + (size_t)jA * HPAD;  // column K=2*hi
      const float* fb1 = fb0 + HPAD;                // column K=2*hi+1
      v2f b;
      b.x = fb0[m16];
      b.y = fb1[m16];
      d0 = __builtin_amdgcn_wmma_f32_16x16x4_f32(false, a, false, b, (short)0, d0,
                                                 false, false);
      b.x = fb0[m16 + 16];
      b.y = fb1[m16 + 16];
      d1 = __builtin_amdgcn_wmma_f32_16x16x4_f32(false, a, false, b, (short)0, d1,
                                                 false, false);
    }
    __syncthreads();
  }

  // --- cross-wave reduction of partial D tiles ---
#pragma unroll
  for (int v = 0; v < 8; ++v) {
    red[w][v * 32 + lane]       = d0[v];
    red[w][(8 + v) * 32 + lane] = d1[v];
  }
  __syncthreads();
  for (int t = tid; t < 512; t += 256) {
    float s = 0.f;
#pragma unroll
    for (int ww = 0; ww < 8; ++ww) s += red[ww][t];
    int half = t >> 8;
    int v    = (t >> 5) & 7;
    int l    = t & 31;
    int m    = v + 8 * (l >> 4);          // C/D layout: VGPR v, lane half -> M
    int n    = (l & 15) + 16 * half;      // lane%16 -> N
    rD[m][n] = s;
  }
  __syncthreads();
  // --- normalize by denominator (col 30), add bias_zero + residual ---
  for (int t = tid; t < TM * HPAD; t += 256) {
    int m = t >> 5, n = t & 31;
    if (n < FDIM) {
      float val = rD[m][n] / rD[m][30] + bz[r * FDIM + n] + x[(i0 + m) * FDIM + n];
      valsWs[((size_t)r * N_NODES + i0 + m) * HPAD + n] = val;
    }
  }
}

// ---------------- Kernel C: out = (sum_r (vals_r @ Wd_r + bd_r)) / 4 ----------------
__global__ void proj_kernel(const float* __restrict__ valsWs,  // [R,N,32]
                            const float* __restrict__ Wd,      // [R,30,32]
                            const float* __restrict__ bd,      // [R,32]
                            float* __restrict__ out) {         // [N,32]
  int idx = blockIdx.x * blockDim.x + threadIdx.x;
  if (idx >= N_NODES * 32) return;
  int n = idx >> 5, e = idx & 31;
  float acc = 0.f;
#pragma unroll
  for (int r = 0; r < RREL; ++r) {
    const float* vr = valsWs + ((size_t)r * N_NODES + n) * HPAD;
    const float* wr = Wd + r * FDIM * 32;
    float a = bd[r * 32 + e];
#pragma unroll
    for (int h = 0; h < FDIM; ++h) a = fmaf(vr[h], wr[h * 32 + e], a);
    acc += a;
  }
  out[idx] = acc * 0.25f;
}

extern "C" void kernel_launch(void* const* d_in, const int* in_sizes, int n_in,
                              void* d_out, int out_size, void* d_ws, size_t ws_size,
                              hipStream_t stream) {
  const float* seq  = (const float*)d_in[0];
  const float* bias = (const float*)d_in[1];
  const float* W    = (const float*)d_in[2];
  const float* a1   = (const float*)d_in[3];
  const float* b1   = (const float*)d_in[4];
  const float* a2   = (const float*)d_in[5];
  const float* b2   = (const float*)d_in[6];
  const float* bz   = (const float*)d_in[7];
  const float* Wd   = (const float*)d_in[8];
  const float* bd   = (const float*)d_in[9];
  float* out = (float*)d_out;

  float* ftsWs  = (float*)d_ws;                                   // R*N*32
  float* valsWs = ftsWs + (size_t)RREL * N_NODES * HPAD;          // R*N*32
  float* f1Ws   = valsWs + (size_t)RREL * N_NODES * HPAD;         // R*N
  float* f2Ws   = f1Ws + (size_t)RREL * N_NODES;                  // R*N

  prep_kernel<<<(RREL * N_NODES + 255) / 256, 256, 0, stream>>>(
      seq, W, a1, b1, a2, b2, ftsWs, f1Ws, f2Ws);
  dim3 g(N_NODES / TM, RREL);
  attn_kernel<<<g, 256, 0, stream>>>(bias, seq, bz, ftsWs, f1Ws, f2Ws, valsWs);
  proj_kernel<<<(N_NODES * 32 + 255) / 256, 256, 0, stream>>>(valsWs, Wd, bd, out);
}